// DimensionalAttentionMask_27238682591388
// MI455X (gfx1250) — compile-verified
//
#include <hip/hip_runtime.h>
#include <hip/hip_bf16.h>

typedef __attribute__((ext_vector_type(2))) float v2f;
typedef __attribute__((ext_vector_type(8))) float v8f;

constexpr int Tn   = 4096;   // sequence length
constexpr int Bn   = 4;      // batches
constexpr int Cn   = 8;      // categories
constexpr int ROWS = Bn * Tn;       // 16384 token rows
constexpr int TJ   = Tn / 16;       // 256 tiles along each output dim
constexpr int JT   = 4;             // j-tiles per wave

// ---------------------------------------------------------------------------
// Kernel 1: P = softmax(emb[tok]), A = P @ compat   (both (B*T, 8) f32)
// ---------------------------------------------------------------------------
__global__ void dam_probs_kernel(const int* __restrict__ tok,
                                 const float* __restrict__ emb,
                                 const float* __restrict__ comp,
                                 float* __restrict__ P,
                                 float* __restrict__ A) {
    int r = blockIdx.x * blockDim.x + threadIdx.x;
    if (r >= ROWS) return;
    const float* row = emb + (size_t)tok[r] * Cn;

    float e[Cn];
    float m = -3.4e38f;
#pragma unroll
    for (int c = 0; c < Cn; ++c) { e[c] = row[c]; m = fmaxf(m, e[c]); }
    float s = 0.f;
#pragma unroll
    for (int c = 0; c < Cn; ++c) { e[c] = __expf(e[c] - m); s += e[c]; }
    float inv = __builtin_amdgcn_rcpf(s);

    float p[Cn];
#pragma unroll
    for (int c = 0; c < Cn; ++c) p[c] = e[c] * inv;

#pragma unroll
    for (int c = 0; c < Cn; ++c) P[(size_t)r * Cn + c] = p[c];

#pragma unroll
    for (int d = 0; d < Cn; ++d) {
        float acc = 0.f;
#pragma unroll
        for (int c = 0; c < Cn; ++c) acc = fmaf(p[c], comp[c * Cn + d], acc);
        A[(size_t)r * Cn + d] = acc;
    }
}

// ---------------------------------------------------------------------------
// Kernel 2: out[b] = sigmoid(A[b] @ P[b]^T) * 2 - 1 via V_WMMA_F32_16X16X4_F32
// One wave computes JT consecutive 16x16 tiles (shared A fragment).
// ---------------------------------------------------------------------------
__global__ void dam_wmma_kernel(const float* __restrict__ A,
                                const float* __restrict__ P,
                                float* __restrict__ out) {
    const int wavesPerBlock = blockDim.x >> 5;
    const int task = blockIdx.x * wavesPerBlock + (threadIdx.x >> 5);
    // tasks per batch: TJ * (TJ/JT) = 256 * 64 = 16384
    const int b   = task >> 14;
    const int rem = task & 16383;
    const int i   = rem >> 6;            // 0..255 : row tile
    const int j0  = (rem & 63) * JT;     // 0..252 : first col tile

    const int lane = threadIdx.x & 31;
    const int hi   = lane >> 4;          // 0: K/M-low half, 1: high half
    const int l    = lane & 15;

    // A fragment: lane holds row (i*16 + l), K pair {2*hi, 2*hi+1} (call 0)
    // and {4+2*hi, 5+2*hi} (call 1).
    const v2f* Arow = (const v2f*)(A + ((size_t)(b * Tn + i * 16 + l)) * Cn);
    const v2f a0 = Arow[hi];
    const v2f a1 = Arow[2 + hi];

#pragma unroll
    for (int jt = 0; jt < JT; ++jt) {
        const int j = j0 + jt;
        // B fragment = P^T: lane holds col (j*16 + l), same K pairing as A.
        const v2f* Prow = (const v2f*)(P + ((size_t)(b * Tn + j * 16 + l)) * Cn);
        const v2f b0 = Prow[hi];
        const v2f b1 = Prow[2 + hi];

        v8f c = {};
        c = __builtin_amdgcn_wmma_f32_16x16x4_f32(false, a0, false, b0,
                                                  (short)0, c, false, false);
        c = __builtin_amdgcn_wmma_f32_16x16x4_f32(false, a1, false, b1,
                                                  (short)0, c, false, false);

        // C/D layout: VGPR v -> row (v + 8*hi), col l
        float* obase = out + ((size_t)(b * Tn + i * 16 + hi * 8)) * Tn + j * 16 + l;
#pragma unroll
        for (int v = 0; v < 8; ++v) {
            float x = c[v];
            // sigmoid(x)*2 - 1  == 2/(1+e^{-x}) - 1
            float e   = __builtin_amdgcn_exp2f(x * -1.4426950408889634f);
            float val = 2.0f * __builtin_amdgcn_rcpf(1.0f + e) - 1.0f;
            __builtin_nontemporal_store(val, obase + (size_t)v * Tn);
        }
    }
}

extern "C" void kernel_launch(void* const* d_in, const int* in_sizes, int n_in,
                              void* d_out, int out_size, void* d_ws, size_t ws_size,
                              hipStream_t stream) {
    const int*   tok  = (const int*)d_in[0];
    const float* emb  = (const float*)d_in[1];
    const float* comp = (const float*)d_in[2];
    float*       out  = (float*)d_out;

    float* Pws = (float*)d_ws;                       // ROWS*8 floats = 512 KB
    float* Aws = Pws + (size_t)ROWS * Cn;            // ROWS*8 floats = 512 KB

    dam_probs_kernel<<<ROWS / 256, 256, 0, stream>>>(tok, emb, comp, Pws, Aws);

    // 262144 tiles / JT per wave = 65536 wave tasks; 8 waves per 256-thr block
    const int tasks  = Bn * TJ * (TJ / JT);
    dam_wmma_kernel<<<tasks / 8, 256, 0, stream>>>(Aws, Pws, out);
}